// GRULayer_37684043055240
// MI455X (gfx1250) — compile-verified
//
#include <hip/hip_runtime.h>
#include <cstdint>
#include <cstddef>

typedef __attribute__((ext_vector_type(2))) float v2f;
typedef __attribute__((ext_vector_type(8))) float v8f;

#define B_    64
#define T_    1024
#define D_    512
#define U_    512
#define G3    1536
#define NBLK  32
#define TPB   512          // 16 waves: K split across wave pairs -> 4 waves/SIMD
#define KSPL  (D_ / 2)     // 256 per K-half
#define PITCH (D_ + 2)     // LDS pitch for weight slices (bank de-clustering)
#define COLS  48           // 3 gates * 16 columns owned per block

// ---------------------------------------------------------------------------
// init: h0 = 0, barrier counter = 0
// ---------------------------------------------------------------------------
__global__ void gru_init_kernel(float* __restrict__ hbuf, unsigned* __restrict__ cnt) {
  int gid = blockIdx.x * blockDim.x + threadIdx.x;
  if (gid < B_ * U_) hbuf[gid] = 0.0f;
  if (gid == 0) *cnt = 0u;
}

__device__ __forceinline__ float sigmoidf_(float v) {
  return 1.0f / (1.0f + __expf(-v));
}

// overflow-safe fast tanh via v_exp
__device__ __forceinline__ float tanhf_fast(float v) {
  float e = __expf(-2.0f * fabsf(v));
  float r = (1.0f - e) / (1.0f + e);
  return copysignf(r, v);
}

// ---------------------------------------------------------------------------
// Fused persistent GRU kernel.
//  - block b owns output columns [16b, 16b+16)
//  - waves 0..7  : xm = x_t @ W   (m-tile = wid&3, K-half = wid>>2)
//  - waves 8..15 : hm = h   @ R   (same split)
//  - W/R column slices live in LDS for the whole kernel (loaded once)
//  - per-step grid barrier via monotonic atomic counter
// ---------------------------------------------------------------------------
__global__ void __launch_bounds__(TPB, 1)
gru_fused_kernel(const float* __restrict__ x,      // [B, T, D]
                 const float* __restrict__ wk,     // [D, 3U]
                 const float* __restrict__ rk,     // [U, 3U]
                 const float* __restrict__ bias,   // [2, 3U]
                 float* __restrict__ hbuf,         // 2 * B*U double buffer
                 unsigned* __restrict__ cnt,
                 float* __restrict__ out) {        // [B, U]
  extern __shared__ float smem[];
  float* Ws   = smem;                     // COLS * PITCH
  float* Rs   = Ws + COLS * PITCH;        // COLS * PITCH
  float* XMB0 = Rs + COLS * PITCH;        // B_ * COLS (xm partial, K-half 0)
  float* XMB1 = XMB0 + B_ * COLS;         // B_ * COLS (xm partial, K-half 1)
  float* HMB0 = XMB1 + B_ * COLS;         // B_ * COLS (hm partial, K-half 0)
  float* HMB1 = HMB0 + B_ * COLS;         // B_ * COLS (hm partial, K-half 1)

  const int tid = threadIdx.x;
  const int j0  = blockIdx.x * 16;        // base output column of this block

  // ---- one-time preload of W/R slices, column-major with pitch ----
  for (int idx = tid; idx < COLS * D_; idx += TPB) {
    int col = idx / D_;                   // 0..47 == g*16 + n
    int k   = idx - col * D_;
    int g   = col >> 4;
    int n   = col & 15;
    int ga  = k * G3 + g * U_ + j0 + n;
    Ws[col * PITCH + k] = wk[ga];
    Rs[col * PITCH + k] = rk[ga];
  }
  __syncthreads();

  const int  wid  = tid >> 5;             // wave id 0..15 (wave32)
  const int  lane = tid & 31;
  const int  half = lane >> 4;            // K split within the wave
  const int  mr   = lane & 15;            // A-matrix row within tile
  const int  nc   = lane & 15;            // B/C/D column within tile
  const bool isX  = (wid < 8);
  const int  widr = wid & 7;
  const int  m0   = (widr & 3) * 16;      // m-tile
  const int  kh   = widr >> 2;            // K-half (0/1)
  const int  kb   = kh * KSPL;
  const int  mrow = m0 + mr;              // batch row for the A operand

  const float* Ls = isX ? Ws : Rs;
  const float* Lz = Ls + (0 * 16 + nc) * PITCH + kb;
  const float* Lr = Ls + (1 * 16 + nc) * PITCH + kb;
  const float* Lh = Ls + (2 * 16 + nc) * PITCH + kb;
  float* obuf = (isX ? XMB0 : HMB0) + kh * (B_ * COLS);

  // ---- hoisted biases for the gate phase (this thread's two columns) ----
  // z,r gates: combined b_i + b_r; h gate: keep b_i_h / b_r_h separate
  const int  c0 = tid >> 6;               // gate-phase col for q=0 (0..7)
  const int  c1 = c0 + 8;                 // gate-phase col for q=1 (8..15)
  const float bz0 = bias[0 * U_ + j0 + c0] + bias[G3 + 0 * U_ + j0 + c0];
  const float bz1 = bias[0 * U_ + j0 + c1] + bias[G3 + 0 * U_ + j0 + c1];
  const float br0 = bias[1 * U_ + j0 + c0] + bias[G3 + 1 * U_ + j0 + c0];
  const float br1 = bias[1 * U_ + j0 + c1] + bias[G3 + 1 * U_ + j0 + c1];
  const float bxh0 = bias[2 * U_ + j0 + c0];
  const float bxh1 = bias[2 * U_ + j0 + c1];
  const float bhh0 = bias[G3 + 2 * U_ + j0 + c0];
  const float bhh1 = bias[G3 + 2 * U_ + j0 + c1];

  const v8f vzero = {0.f, 0.f, 0.f, 0.f, 0.f, 0.f, 0.f, 0.f};

  for (int t = 0; t < T_; ++t) {
    const float* hcur  = hbuf + (size_t)(t & 1) * (B_ * U_);
    float*       hnext = hbuf + (size_t)((t + 1) & 1) * (B_ * U_);

    // ---- GEMM phase: 16x16 tile per gate over this wave's K-half ----
    const float* aptr = (isX ? (x + ((size_t)mrow * T_ + t) * D_)
                             : (hcur + (size_t)mrow * U_)) + kb;
    v8f az = vzero, ar = vzero, ah = vzero;
    for (int k = 0; k < KSPL; k += 4) {
      const int kk = k + half * 2;
      v2f a  = *(const v2f*)(aptr + kk);
      v2f b0 = *(const v2f*)(Lz + kk);
      v2f b1 = *(const v2f*)(Lr + kk);
      v2f b2 = *(const v2f*)(Lh + kk);
      az = __builtin_amdgcn_wmma_f32_16x16x4_f32(false, a, false, b0, (short)0, az, false, false);
      ar = __builtin_amdgcn_wmma_f32_16x16x4_f32(false, a, false, b1, (short)0, ar, false, false);
      ah = __builtin_amdgcn_wmma_f32_16x16x4_f32(false, a, false, b2, (short)0, ah, false, false);
    }

    // ---- stage raw partial accumulators into LDS: obuf[row][g*16+nc] ----
#pragma unroll
    for (int i = 0; i < 8; ++i) {
      int row = m0 + half * 8 + i;        // C/D layout: row = 8*half + i
      obuf[row * COLS + 0 * 16 + nc] = az[i];
      obuf[row * COLS + 1 * 16 + nc] = ar[i];
      obuf[row * COLS + 2 * 16 + nc] = ah[i];
    }
    __syncthreads();

    // ---- gate phase: 64 rows x 16 cols over 512 threads (2 elems each) ----
#pragma unroll
    for (int q = 0; q < 2; ++q) {
      int idx = tid + q * TPB;
      int row = idx & 63;
      int col = idx >> 6;
      int base = row * COLS + col;
      float xz = XMB0[base]      + XMB1[base];
      float hz = HMB0[base]      + HMB1[base];
      float xr = XMB0[base + 16] + XMB1[base + 16];
      float hr = HMB0[base + 16] + HMB1[base + 16];
      float xh = XMB0[base + 32] + XMB1[base + 32];
      float hh = HMB0[base + 32] + HMB1[base + 32];
      float bz = q ? bz1 : bz0, br = q ? br1 : br0;
      float bxh = q ? bxh1 : bxh0, bhh = q ? bhh1 : bhh0;
      float hprev = hcur[row * U_ + j0 + col];
      float z  = sigmoidf_(xz + hz + bz);
      float r  = sigmoidf_(xr + hr + br);
      float hc = tanhf_fast(xh + bxh + r * (hh + bhh));
      float hn = z * hprev + (1.0f - z) * hc;
      hnext[row * U_ + j0 + col] = hn;
      if (t == T_ - 1) out[row * U_ + j0 + col] = hn;
    }

    // ---- grid-wide barrier (monotonic counter; 32 co-resident blocks) ----
    __threadfence();
    __syncthreads();
    if (tid == 0) {
      atomicAdd(cnt, 1u);
      const unsigned target = (unsigned)(t + 1) * NBLK;
      while (atomicAdd(cnt, 0u) < target) {
        __builtin_amdgcn_s_sleep(1);
      }
    }
    __syncthreads();
  }
}

// ---------------------------------------------------------------------------
extern "C" void kernel_launch(void* const* d_in, const int* in_sizes, int n_in,
                              void* d_out, int out_size, void* d_ws, size_t ws_size,
                              hipStream_t stream) {
  const float* x    = (const float*)d_in[0];   // [B, T, D]
  const float* wk   = (const float*)d_in[1];   // [D, 3U]
  const float* rk   = (const float*)d_in[2];   // [U, 3U]
  const float* bias = (const float*)d_in[3];   // [2, 3U]
  float* out = (float*)d_out;                  // [B, U]

  // workspace: h double buffer + barrier counter
  float*    hbuf = (float*)d_ws;                                  // 2*B*U floats
  unsigned* cnt  = (unsigned*)((char*)d_ws + (size_t)2 * B_ * U_ * sizeof(float));

  gru_init_kernel<<<(B_ * U_ + 255) / 256, 256, 0, stream>>>(hbuf, cnt);

  const size_t shbytes = (size_t)(2 * COLS * PITCH + 4 * B_ * COLS) * sizeof(float);
  (void)hipFuncSetAttribute((const void*)gru_fused_kernel,
                            hipFuncAttributeMaxDynamicSharedMemorySize,
                            (int)shbytes);
  gru_fused_kernel<<<NBLK, TPB, shbytes, stream>>>(x, wk, rk, bias, hbuf, cnt, out);
}